// SoftTargetEntmaxBisectLoss_32710470926798
// MI455X (gfx1250) — compile-verified
//
#include <hip/hip_runtime.h>

// ---------------------------------------------------------------------------
// SoftTargetEntmaxBisectLoss for MI455X (gfx1250, wave32)
//   logits, targets: [4096, 32000] fp32 ; output: scalar fp32 mean loss
//
// Strategy: one 512-thread block per row. Row logits live in VGPRs
// (64 floats/thread), so the 50 bisection passes never touch memory.
// Targets are DMA-prefetched into LDS with gfx1250 async-to-LDS copies,
// overlapped with the bisection, consumed only in the epilogue.
// Block reductions use a single barrier per call (alternating LDS buffers).
// ---------------------------------------------------------------------------

constexpr int B_ROWS = 4096;
constexpr int V      = 32000;
constexpr int V4     = V / 4;                  // 8000 float4 per row
constexpr int TPB    = 512;                    // 16 waves (wave32)
constexpr int NCHUNK = (V4 + TPB - 1) / TPB;   // 16 chunks of float4
constexpr int NELEM  = NCHUNK * 4;             // 64 floats per thread
constexpr int NWAVE  = TPB / 32;               // 16

#define USE_ASYNC_LDS 1

__device__ __forceinline__ float wave_sum(float v) {
#pragma unroll
  for (int m = 16; m >= 1; m >>= 1) v += __shfl_xor(v, m, 32);
  return v;
}
__device__ __forceinline__ float wave_max(float v) {
#pragma unroll
  for (int m = 16; m >= 1; m >>= 1) v = fmaxf(v, __shfl_xor(v, m, 32));
  return v;
}

// Single-barrier deterministic block reduction with uniform broadcast.
// Callers must alternate `slot` (0/1) between consecutive calls; the one
// barrier per call then guarantees no WAR across the 2-call reuse distance.
template <bool IS_MAX>
__device__ __forceinline__ float block_reduce(float v, float (&red)[2][NWAVE],
                                              int slot) {
  const int lane = threadIdx.x & 31;
  const int wave = threadIdx.x >> 5;
  v = IS_MAX ? wave_max(v) : wave_sum(v);
  if (lane == 0) red[slot][wave] = v;
  __syncthreads();
  // Every thread re-reads all 16 wave partials (4x ds_load_b128, LDS
  // broadcast) and combines them in a fixed order -> identical result in
  // every thread, deterministic across runs.
  const float4* r4 = (const float4*)&red[slot][0];
  const float4 a = r4[0], b = r4[1], c = r4[2], d = r4[3];
  if (IS_MAX) {
    float m0 = fmaxf(fmaxf(a.x, a.y), fmaxf(a.z, a.w));
    float m1 = fmaxf(fmaxf(b.x, b.y), fmaxf(b.z, b.w));
    float m2 = fmaxf(fmaxf(c.x, c.y), fmaxf(c.z, c.w));
    float m3 = fmaxf(fmaxf(d.x, d.y), fmaxf(d.z, d.w));
    return fmaxf(fmaxf(m0, m1), fmaxf(m2, m3));
  } else {
    float s0 = (a.x + a.y) + (a.z + a.w);
    float s1 = (b.x + b.y) + (b.z + b.w);
    float s2 = (c.x + c.y) + (c.z + c.w);
    float s3 = (d.x + d.y) + (d.z + d.w);
    return (s0 + s1) + (s2 + s3);
  }
}

__global__ __launch_bounds__(TPB)
void entmax_row_kernel(const float* __restrict__ logits,
                       const float* __restrict__ targets,
                       float* __restrict__ row_loss) {
  __shared__ __align__(16) float red[2][NWAVE];
  __shared__ float4 tbuf[NCHUNK * TPB];        // 128 KB target stage (LDS: 320KB/WGP)

  const int row = blockIdx.x;
  const int tid = threadIdx.x;
  const float4* __restrict__ lrow = (const float4*)(logits  + (size_t)row * V);
  const float4* __restrict__ trow = (const float4*)(targets + (size_t)row * V);

#if USE_ASYNC_LDS
  // gfx1250 async DMA: stream the whole targets row into LDS now; it completes
  // in the shadow of the 50-iteration bisection. Pad lanes clamp their address
  // in-bounds (their staged data is discarded at consume time). Each lane
  // later reads back only its own slot, so s_wait_asynccnt alone suffices.
#pragma unroll
  for (int c = 0; c < NCHUNK; ++c) {
    const int idx4 = c * TPB + tid;
    const int src  = (idx4 < V4) ? idx4 : (V4 - 1);
    unsigned long long ga = (unsigned long long)(const void*)(trow + src);
    unsigned int       la = (unsigned int)(unsigned long long)(const void*)(&tbuf[c * TPB + tid]);
    asm volatile("global_load_async_to_lds_b128 %0, %1, off"
                 :: "v"(la), "v"(ga) : "memory");
  }
#endif

  // Load the logits row into registers, pre-scaled by (alpha-1)=0.5 : Xs = 0.5*X
  float xs[NELEM];
#pragma unroll
  for (int c = 0; c < NCHUNK; ++c) {
    const int idx4 = c * TPB + tid;
    float4 x;
    if (idx4 < V4) x = lrow[idx4];
    else           x = make_float4(-1.0e30f, -1.0e30f, -1.0e30f, -1.0e30f);
    xs[c * 4 + 0] = 0.5f * x.x;
    xs[c * 4 + 1] = 0.5f * x.y;
    xs[c * 4 + 2] = 0.5f * x.z;
    xs[c * 4 + 3] = 0.5f * x.w;
  }

  // max over the row (reduction slot 0)
  float mv = -3.0e38f;
#pragma unroll
  for (int i = 0; i < NELEM; ++i) mv = fmaxf(mv, xs[i]);
  const float max_val = block_reduce<true>(mv, red, 0);

  float       tau_lo = max_val - 1.0f;
  const float tau_hi = max_val - 0.005590169943749474f;  // (1/32000)^(alpha-1)

  // f_lo (fixed across iterations, as in the reference) -- slot 1
  float acc = 0.0f;
#pragma unroll
  for (int i = 0; i < NELEM; ++i) {
    const float z = fmaxf(xs[i] - tau_lo, 0.0f);
    acc = fmaf(z, z, acc);
  }
  const float f_lo = block_reduce<false>(acc, red, 1) - 1.0f;

  float dm    = tau_hi - tau_lo;
  float tau_m = tau_lo;

#pragma unroll 1
  for (int it = 0; it < 50; ++it) {          // slots 0,1,0,1,... (prev was 1)
    dm *= 0.5f;
    tau_m = tau_lo + dm;
    float a = 0.0f;
#pragma unroll
    for (int i = 0; i < NELEM; ++i) {
      const float z = fmaxf(xs[i] - tau_m, 0.0f);   // p(tau) = relu(z)^2
      a = fmaf(z, z, a);
    }
    const float f_m = block_reduce<false>(a, red, it & 1) - 1.0f;
    if (f_m * f_lo >= 0.0f) tau_lo = tau_m;          // uniform across the block
  }

  // --- epilogue: S = sum z^2, S3 = sum z^3, SPX = sum z^2*x, STX = sum t*x ---
#if USE_ASYNC_LDS
  asm volatile("s_wait_asynccnt 0" ::: "memory");    // targets now resident in LDS
#endif
  float s2 = 0.0f, s3 = 0.0f, spx = 0.0f, stx = 0.0f;
#pragma unroll
  for (int c = 0; c < NCHUNK; ++c) {
    const int idx4 = c * TPB + tid;
#if USE_ASYNC_LDS
    float4 t = tbuf[c * TPB + tid];
#else
    float4 t = (idx4 < V4) ? trow[idx4] : make_float4(0.f, 0.f, 0.f, 0.f);
#endif
    if (idx4 >= V4) t = make_float4(0.f, 0.f, 0.f, 0.f);
    const float tv[4] = {t.x, t.y, t.z, t.w};
#pragma unroll
    for (int j = 0; j < 4; ++j) {
      const float xv = xs[c * 4 + j];                // = 0.5 * logit, so x = 2*xv
      const float z  = fmaxf(xv - tau_m, 0.0f);
      const float z2 = z * z;
      s2 += z2;
      s3  = fmaf(z2, z,  s3);
      spx = fmaf(z2, xv, spx);
      stx = fmaf(tv[j], xv, stx);
    }
  }
  // last loop slot was (49&1)=1 -> continue alternating 0,1,0,1
  const float S   = block_reduce<false>(s2,  red, 0);
  const float S3  = block_reduce<false>(s3,  red, 1);
  const float SPX = block_reduce<false>(spx, red, 0);
  const float STX = block_reduce<false>(stx, red, 1);

  if (tid == 0) {
    // sum(p_norm^alpha) = S3 / S^1.5 ; omega = (1 - .) / (alpha*(alpha-1)=0.75)
    const float omega = (1.0f - S3 / (S * sqrtf(S))) * (4.0f / 3.0f);
    // sum((p_norm - t) * x) = (2*SPX)/S - 2*STX   (x = 2*xs)
    row_loss[row] = omega + (2.0f * SPX) / S - 2.0f * STX;
  }
}

// Deterministic fixed-tree mean over the 4096 per-row losses.
__global__ __launch_bounds__(1024)
void mean_kernel(const float* __restrict__ rl, float* __restrict__ out) {
  __shared__ float red[32];
  const int tid  = threadIdx.x;
  const int lane = tid & 31;
  const int wave = tid >> 5;
  float v = rl[tid] + rl[tid + 1024] + rl[tid + 2048] + rl[tid + 3072];
  v = wave_sum(v);
  if (lane == 0) red[wave] = v;
  __syncthreads();
  if (wave == 0) {
    float t = red[lane];
    t = wave_sum(t);
    if (lane == 0) out[0] = t * (1.0f / (float)B_ROWS);
  }
}

extern "C" void kernel_launch(void* const* d_in, const int* in_sizes, int n_in,
                              void* d_out, int out_size, void* d_ws, size_t ws_size,
                              hipStream_t stream) {
  (void)in_sizes; (void)n_in; (void)out_size; (void)ws_size;
  const float* logits  = (const float*)d_in[0];
  const float* targets = (const float*)d_in[1];
  float* out = (float*)d_out;
  float* rl  = (float*)d_ws;                       // 4096 floats of scratch

  entmax_row_kernel<<<B_ROWS, TPB, 0, stream>>>(logits, targets, rl);
  mean_kernel<<<1, 1024, 0, stream>>>(rl, out);
}